// BayesPosLinear_52183852646954
// MI455X (gfx1250) — compile-verified
//
#include <hip/hip_runtime.h>

typedef __attribute__((ext_vector_type(2))) float v2f;
typedef __attribute__((ext_vector_type(8))) float v8f;

#define S_DIM   16
#define B_DIM   32
#define IN_DIM  1024
#define OUT_DIM 1024
#define KC      256          // K-chunk staged in LDS per iteration
#define LDW     260          // padded LDS row stride (floats); 260*4B % 16 == 0, kills bank conflicts
#define LOG2E   1.44269504088896340736f

__device__ __forceinline__ float softplus_stable(float x) {
    // log(1+e^x) = max(x,0) + log1p(exp(-|x|))
    return fmaxf(x, 0.0f) + log1pf(expf(-fabsf(x)));
}

// CDNA5 async global->LDS copy, 16B per lane. vdst = 32-bit LDS address,
// vaddr = 64-bit global address, tracked by ASYNCcnt.
__device__ __forceinline__ void async_copy_b128(const void* gptr, void* lds_ptr) {
    unsigned lds_addr = (unsigned)(unsigned long long)(lds_ptr); // low 32b of flat = LDS offset
    asm volatile("global_load_async_to_lds_b128 %0, %1, off"
                 :: "v"(lds_addr), "v"(gptr) : "memory");
}

__device__ __forceinline__ void async_wait0() {
#if __has_builtin(__builtin_amdgcn_s_wait_asynccnt)
    __builtin_amdgcn_s_wait_asynccnt(0);
#else
    asm volatile("s_wait_asynccnt 0x0" ::: "memory");
#endif
}

// ---------------------------------------------------------------------------
// Kernel 1: write interleaved (mu*log2e, (1e-6+softplus(eta))*log2e) pairs to ws
// and produce KL(w) per-block partials. 1024 blocks x 256 thr x 4 elem = OUT*IN.
// ---------------------------------------------------------------------------
__global__ __launch_bounds__(256) void k_prep_w(const float* __restrict__ w_mu,
                                                const float* __restrict__ w_std_eta,
                                                float* __restrict__ ms,      // 2*OUT*IN floats
                                                float* __restrict__ partial) {
    __shared__ float red[256];
    const int tid = threadIdx.x;
    float acc = 0.0f;
#pragma unroll
    for (int j = 0; j < 4; ++j) {
        size_t idx = (size_t)blockIdx.x * 1024 + (size_t)j * 256 + tid;
        float eta = w_std_eta[idx];
        float sd  = 1e-6f + softplus_stable(eta);
        float mu  = w_mu[idx];
        float2 p;
        p.x = mu * LOG2E;
        p.y = sd * LOG2E;
        *(float2*)(ms + 2 * idx) = p;
        acc += -logf(sd) + 0.5f * (sd * sd + mu * mu) - 0.5f;
    }
    red[tid] = acc;
    __syncthreads();
    for (int s = 128; s > 0; s >>= 1) {
        if (tid < s) red[tid] += red[tid + s];
        __syncthreads();
    }
    if (tid == 0) partial[blockIdx.x] = red[0];
}

// ---------------------------------------------------------------------------
// Kernel 2: sum 1024 kl_w partials -> d_out[S*B*OUT]
// ---------------------------------------------------------------------------
__global__ __launch_bounds__(256) void k_sum_partials(const float* __restrict__ partial,
                                                      float* __restrict__ dst) {
    __shared__ float red[256];
    const int tid = threadIdx.x;
    float acc = partial[tid] + partial[tid + 256] + partial[tid + 512] + partial[tid + 768];
    red[tid] = acc;
    __syncthreads();
    for (int s = 128; s > 0; s >>= 1) {
        if (tid < s) red[tid] += red[tid + s];
        __syncthreads();
    }
    if (tid == 0) dst[0] = red[0];
}

// ---------------------------------------------------------------------------
// Kernel 3: std_b (to ws) + KL(b) -> d_out[S*B*OUT + 1]
// ---------------------------------------------------------------------------
__global__ __launch_bounds__(256) void k_std_b_kl(const float* __restrict__ b_mu,
                                                  const float* __restrict__ b_std_eta,
                                                  float* __restrict__ std_b,
                                                  float* __restrict__ dst) {
    __shared__ float red[256];
    const int tid = threadIdx.x;
    float acc = 0.0f;
#pragma unroll
    for (int j = 0; j < 4; ++j) {
        int i = tid + j * 256;
        float sd = 1e-6f + softplus_stable(b_std_eta[i]);
        std_b[i] = sd;
        float mu = b_mu[i];
        acc += -logf(sd) + 0.5f * (sd * sd + mu * mu) - 0.5f;
    }
    red[tid] = acc;
    __syncthreads();
    for (int s = 128; s > 0; s >>= 1) {
        if (tid < s) red[tid] += red[tid + s];
        __syncthreads();
    }
    if (tid == 0) dst[0] = red[0];
}

// ---------------------------------------------------------------------------
// Main kernel: per (b, o_tile) compute out[0:16, b, o0:o0+16] via WMMA f32 16x16x4.
// A tile: async global->LDS. W tile: w = exp2(mu' + std'*eps), one fma + one v_exp
// per element. 4 waves split K; LDS cross-wave reduction + bias at the end.
// ---------------------------------------------------------------------------
__global__ __launch_bounds__(128) void k_gemm(const float* __restrict__ input,
                                              const float* __restrict__ ms,
                                              const float* __restrict__ b_mu,
                                              const float* __restrict__ std_b,
                                              const float* __restrict__ eps_w,
                                              const float* __restrict__ eps_b,
                                              float* __restrict__ out) {
    __shared__ float Ash[S_DIM * LDW];   // input tile  [16][KC]
    __shared__ float Wsh[16 * LDW];      // weight tile [16 o][KC]

    const int b    = blockIdx.y;
    const int o0   = blockIdx.x * 16;
    const int tid  = threadIdx.x;
    const int wave = tid >> 5;
    const int lane = tid & 31;
    const int hh   = lane >> 4;   // half-wave selector (K pairing)
    const int mn   = lane & 15;   // M for A-rows, N for B-cols

    v8f c = {0.f, 0.f, 0.f, 0.f, 0.f, 0.f, 0.f, 0.f};

    for (int i0 = 0; i0 < IN_DIM; i0 += KC) {
        // ---- stage A tile asynchronously: input[s, b, i0:i0+KC] ----
#pragma unroll
        for (int r = 0; r < 8; ++r) {
            int idx  = r * 128 + tid;          // 0..1023
            int row  = idx >> 6;               // s
            int col4 = idx & 63;
            async_copy_b128(input + ((size_t)(row * B_DIM + b)) * IN_DIM + i0 + col4 * 4,
                            &Ash[row * LDW + col4 * 4]);
        }
        // ---- stage W tile: exp2(mu' + std'*eps) for o0..o0+15, i0:i0+KC ----
#pragma unroll
        for (int r = 0; r < 8; ++r) {
            int idx  = r * 128 + tid;
            int row  = idx >> 6;               // o within tile
            int col4 = idx & 63;
            size_t ge = ((size_t)(b * OUT_DIM + o0 + row)) * IN_DIM + i0 + col4 * 4;
            size_t gm = ((size_t)(o0 + row)) * IN_DIM + i0 + col4 * 4;
            float4 e   = *(const float4*)(eps_w + ge);
            float4 ms0 = *(const float4*)(ms + 2 * gm);       // mu0,sd0,mu1,sd1 (pre-scaled)
            float4 ms1 = *(const float4*)(ms + 2 * gm + 4);   // mu2,sd2,mu3,sd3
            float4 w;
            w.x = __builtin_amdgcn_exp2f(fmaf(ms0.y, e.x, ms0.x));
            w.y = __builtin_amdgcn_exp2f(fmaf(ms0.w, e.y, ms0.z));
            w.z = __builtin_amdgcn_exp2f(fmaf(ms1.y, e.z, ms1.x));
            w.w = __builtin_amdgcn_exp2f(fmaf(ms1.w, e.w, ms1.z));
            *(float4*)(&Wsh[row * LDW + col4 * 4]) = w;
        }
        async_wait0();          // our wave's async LDS writes done
        __syncthreads();        // all waves' staging visible

        // ---- WMMA over this chunk: wave handles K sub-range of 64 ----
#pragma unroll
        for (int t = 0; t < 16; ++t) {
            int k0 = wave * 64 + t * 4 + hh * 2;
            v2f a  = *(const v2f*)(&Ash[mn * LDW + k0]);   // {A[m][k], A[m][k+1]}
            v2f bb = *(const v2f*)(&Wsh[mn * LDW + k0]);   // {B[k][n], B[k+1][n]}
            c = __builtin_amdgcn_wmma_f32_16x16x4_f32(
                    /*neg_a=*/false, a, /*neg_b=*/false, bb,
                    /*c_mod=*/(short)0, c, /*reuse_a=*/false, /*reuse_b=*/false);
        }
        __syncthreads();
    }

    // ---- cross-wave reduction of the 4 partial C tiles (reuse Ash) ----
#pragma unroll
    for (int r = 0; r < 8; ++r)
        Ash[wave * 256 + (r + 8 * hh) * 16 + mn] = c[r];
    __syncthreads();

    for (int e = tid; e < 256; e += 128) {
        int m = e >> 4;
        int n = e & 15;
        float sum = Ash[e] + Ash[256 + e] + Ash[512 + e] + Ash[768 + e];
        int o = o0 + n;
        float bias = b_mu[o] + std_b[o] * eps_b[b * OUT_DIM + o];
        out[((size_t)(m * B_DIM + b)) * OUT_DIM + o] = sum + bias;
    }
}

// ---------------------------------------------------------------------------
extern "C" void kernel_launch(void* const* d_in, const int* in_sizes, int n_in,
                              void* d_out, int out_size, void* d_ws, size_t ws_size,
                              hipStream_t stream) {
    (void)in_sizes; (void)n_in; (void)out_size; (void)ws_size;
    const float* input     = (const float*)d_in[0];
    const float* w_mu      = (const float*)d_in[1];
    const float* w_std_eta = (const float*)d_in[2];
    const float* b_mu      = (const float*)d_in[3];
    const float* b_std_eta = (const float*)d_in[4];
    const float* eps_w     = (const float*)d_in[5];
    const float* eps_b     = (const float*)d_in[6];
    float* out = (float*)d_out;

    float* ws      = (float*)d_ws;
    float* ms      = ws;                                      // 2*OUT*IN floats (8 MB)
    float* std_b   = ms + (size_t)2 * OUT_DIM * IN_DIM;       // OUT floats
    float* partial = std_b + OUT_DIM;                         // 1024 floats

    const size_t out_elems = (size_t)S_DIM * B_DIM * OUT_DIM;  // 524288

    k_prep_w<<<1024, 256, 0, stream>>>(w_mu, w_std_eta, ms, partial);
    k_sum_partials<<<1, 256, 0, stream>>>(partial, out + out_elems);        // kl_w
    k_std_b_kl<<<1, 256, 0, stream>>>(b_mu, b_std_eta, std_b, out + out_elems + 1);  // kl_b

    dim3 grid(OUT_DIM / 16, B_DIM);
    k_gemm<<<grid, 128, 0, stream>>>(input, ms, b_mu, std_b, eps_w, eps_b, out);
}